// LightGCN_80556406604525
// MI455X (gfx1250) — compile-verified
//
#include <hip/hip_runtime.h>
#include <math.h>

#define THREADS 256

// ============ CSR construction ============

__global__ void zero_u32(unsigned* __restrict__ p, int n) {
  int i = blockIdx.x * blockDim.x + threadIdx.x;
  if (i < n) p[i] = 0u;
}

// cnt[dst[e]]++  (native global_atomic_add_u32)
__global__ void count_deg(const int* __restrict__ dst, unsigned* __restrict__ cnt, int E) {
  int e = blockIdx.x * blockDim.x + threadIdx.x;
  if (e < E) atomicAdd(&cnt[dst[e]], 1u);
}

// dis[i] = cnt>0 ? rsqrt(cnt) : 0
__global__ void dis_from_cnt(const unsigned* __restrict__ cnt, float* __restrict__ dis, int N) {
  int i = blockIdx.x * blockDim.x + threadIdx.x;
  if (i < N) {
    unsigned c = cnt[i];
    dis[i] = c ? rsqrtf((float)c) : 0.0f;
  }
}

// per-256-block sum of cnt -> bs[block]
__global__ void block_reduce(const unsigned* __restrict__ cnt, unsigned* __restrict__ bs, int N) {
  __shared__ unsigned sh[THREADS];
  int i = blockIdx.x * THREADS + threadIdx.x;
  sh[threadIdx.x] = (i < N) ? cnt[i] : 0u;
  __syncthreads();
  for (int off = THREADS / 2; off > 0; off >>= 1) {
    if (threadIdx.x < off) sh[threadIdx.x] += sh[threadIdx.x + off];
    __syncthreads();
  }
  if (threadIdx.x == 0) bs[blockIdx.x] = sh[0];
}

// single-block exclusive scan of bs[0..nb) in place (chunks of 1024, running offset)
__global__ void scan_block_sums(unsigned* __restrict__ bs, int nb) {
  __shared__ unsigned sh[1024];
  unsigned run = 0;
  for (int base = 0; base < nb; base += 1024) {
    int i = base + (int)threadIdx.x;
    unsigned v = (i < nb) ? bs[i] : 0u;
    sh[threadIdx.x] = v;
    __syncthreads();
    for (int off = 1; off < 1024; off <<= 1) {
      unsigned t = (threadIdx.x >= (unsigned)off) ? sh[threadIdx.x - off] : 0u;
      __syncthreads();
      sh[threadIdx.x] += t;
      __syncthreads();
    }
    unsigned incl = sh[threadIdx.x];
    if (i < nb) bs[i] = run + (incl - v);   // exclusive
    unsigned total = sh[1023];
    __syncthreads();                         // protect sh before next chunk writes
    run += total;
  }
}

// rowptr[i] = bs[block] + local exclusive scan; cursor[i] = rowptr[i]; rowptr[N] = E
__global__ void write_rowptr(const unsigned* __restrict__ cnt, const unsigned* __restrict__ bs,
                             unsigned* __restrict__ rowptr, unsigned* __restrict__ cursor,
                             int N, unsigned E) {
  __shared__ unsigned sh[THREADS];
  int i = blockIdx.x * THREADS + threadIdx.x;
  unsigned v = (i < N) ? cnt[i] : 0u;
  sh[threadIdx.x] = v;
  __syncthreads();
  for (int off = 1; off < THREADS; off <<= 1) {
    unsigned t = (threadIdx.x >= (unsigned)off) ? sh[threadIdx.x - off] : 0u;
    __syncthreads();
    sh[threadIdx.x] += t;
    __syncthreads();
  }
  unsigned excl = sh[threadIdx.x] - v;
  if (i < N) {
    unsigned r = bs[blockIdx.x] + excl;
    rowptr[i] = r;
    cursor[i] = r;
  }
  if (i == 0) rowptr[N] = E;
}

// slot = cursor[dst]++ ; csr_src[slot] = src ; csr_w[slot] = dis[src]*dis[dst]
__global__ void fill_csr(const int* __restrict__ src, const int* __restrict__ dst,
                         const float* __restrict__ dis, unsigned* __restrict__ cursor,
                         int* __restrict__ csrc, float* __restrict__ cw, int E) {
  int e = blockIdx.x * blockDim.x + threadIdx.x;
  if (e < E) {
    int s = src[e], d = dst[e];
    unsigned slot = atomicAdd(&cursor[d], 1u);   // global_atomic_add_u32 rtn
    csrc[slot] = s;
    cw[slot] = dis[s] * dis[d];
  }
}

// ============ dense init / propagation ============

// x0 = concat(user, item); seed output accumulator with x0
__global__ void init_x(const float* __restrict__ uw, const float* __restrict__ iw,
                       float* __restrict__ x, float* __restrict__ out,
                       size_t u64, size_t n64) {
  size_t i = (size_t)blockIdx.x * blockDim.x + threadIdx.x;
  if (i < n64) {
    float v = (i < u64) ? uw[i] : iw[i - u64];
    x[i] = v;
    out[i] = v;
  }
}

// One 16-lane group owns destination row d (2 rows per wave32), no atomics.
// Edges are consumed in batches of 16: lane L loads CSR edge base+L (coalesced)
// and immediately prefetches its source row (2x global_prefetch_b8 covering
// the 256B row, locality=RT so it stays in the 192MB L2). The MAC loop then
// broadcasts (src, w) from lane t via __shfl (LDS-permute, no memory traffic)
// and does a coalesced 256B b128 row load + v_fmac. With avg degree ~8, one
// batch covers the whole row, so every gather is prefetched before first use.
// Fused epilogue: xn[d] = acc ; out[d] = (out[d] + acc) * scale.
__global__ void gather_step(const float4* __restrict__ x, float4* __restrict__ xn,
                            float4* __restrict__ out,
                            const unsigned* __restrict__ rowptr,
                            const int* __restrict__ csrc, const float* __restrict__ cw,
                            int N, float scale) {
  int gid  = blockIdx.x * blockDim.x + threadIdx.x;
  int d    = gid >> 4;
  int lane = gid & 15;
  if (d >= N) return;

  unsigned beg = rowptr[d];
  unsigned end = rowptr[d + 1];
  float4 acc = make_float4(0.f, 0.f, 0.f, 0.f);

  for (unsigned base = beg; base < end; base += 16u) {
    unsigned j   = base + (unsigned)lane;
    int      myS = 0;
    float    myW = 0.0f;
    if (j < end) {
      myS = csrc[j];                         // coalesced edge fetch
      myW = cw[j];
      const char* row = (const char*)(x + (size_t)myS * 16);
      __builtin_prefetch((const void*)row,         0, 3);   // global_prefetch_b8
      __builtin_prefetch((const void*)(row + 128), 0, 3);
    }
    unsigned rem = end - base;
    int nbatch = (rem < 16u) ? (int)rem : 16;
    for (int t = 0; t < nbatch; ++t) {
      int   s = __shfl(myS, t, 16);          // broadcast within 16-lane group
      float w = __shfl(myW, t, 16);
      float4 v = x[(size_t)s * 16 + lane];   // global_load_b128, coalesced
      acc.x = fmaf(w, v.x, acc.x);
      acc.y = fmaf(w, v.y, acc.y);
      acc.z = fmaf(w, v.z, acc.z);
      acc.w = fmaf(w, v.w, acc.w);
    }
  }

  size_t o = (size_t)d * 16 + lane;
  xn[o] = acc;
  float4 ov = out[o];
  ov.x = (ov.x + acc.x) * scale;
  ov.y = (ov.y + acc.y) * scale;
  ov.z = (ov.z + acc.z) * scale;
  ov.w = (ov.w + acc.w) * scale;
  out[o] = ov;
}

// ============ launcher ============

extern "C" void kernel_launch(void* const* d_in, const int* in_sizes, int n_in,
                              void* d_out, int out_size, void* d_ws, size_t ws_size,
                              hipStream_t stream) {
  (void)n_in; (void)out_size; (void)ws_size;

  const float* uw  = (const float*)d_in[0];   // [U,64] f32
  const float* iw  = (const float*)d_in[1];   // [I,64] f32
  const int*   ei  = (const int*)d_in[2];     // [2,E]  int32 (row0=src, row1=dst)
  float*       out = (float*)d_out;           // [N,64] f32

  const size_t u64 = (size_t)in_sizes[0];
  const size_t i64 = (size_t)in_sizes[1];
  const size_t n64 = u64 + i64;               // N*64 floats
  const int    N   = (int)(n64 / 64);
  const int    E   = in_sizes[2] / 2;
  const int*   src = ei;
  const int*   dst = ei + E;
  const int    nb  = (N + THREADS - 1) / THREADS;

  // workspace carve-up, 256B aligned
  auto a256 = [](size_t x) { return (x + 255) & ~(size_t)255; };
  char*     ws     = (char*)d_ws;
  size_t    off    = 0;
  unsigned* cnt    = (unsigned*)(ws + off); off = a256(off + (size_t)N * 4);
  float*    dis    = (float*)   (ws + off); off = a256(off + (size_t)N * 4);
  unsigned* rowptr = (unsigned*)(ws + off); off = a256(off + (size_t)(N + 1) * 4);
  unsigned* cursor = (unsigned*)(ws + off); off = a256(off + (size_t)N * 4);
  unsigned* bs     = (unsigned*)(ws + off); off = a256(off + (size_t)nb * 4);
  int*      csrc   = (int*)     (ws + off); off = a256(off + (size_t)E * 4);
  float*    cw     = (float*)   (ws + off); off = a256(off + (size_t)E * 4);
  float*    xA     = (float*)   (ws + off); off = a256(off + n64 * 4);
  float*    xB     = (float*)   (ws + off); off = a256(off + n64 * 4);

  const int bE   = (E + THREADS - 1) / THREADS;
  const int bN64 = (int)((n64 + THREADS - 1) / THREADS);

  // ---- CSR build (u32 atomics only) ----
  zero_u32     <<<nb, THREADS, 0, stream>>>(cnt, N);
  count_deg    <<<bE, THREADS, 0, stream>>>(dst, cnt, E);
  dis_from_cnt <<<nb, THREADS, 0, stream>>>(cnt, dis, N);
  block_reduce <<<nb, THREADS, 0, stream>>>(cnt, bs, N);
  scan_block_sums<<<1, 1024, 0, stream>>>(bs, nb);
  write_rowptr <<<nb, THREADS, 0, stream>>>(cnt, bs, rowptr, cursor, N, (unsigned)E);
  fill_csr     <<<bE, THREADS, 0, stream>>>(src, dst, dis, cursor, csrc, cw, E);

  // ---- x0 + output seed ----
  init_x<<<bN64, THREADS, 0, stream>>>(uw, iw, xA, out, u64, n64);

  // ---- 3 propagation steps, atomic-free gather, fused accumulate ----
  const int gthreads = N * 16;
  const int gblocks  = (gthreads + THREADS - 1) / THREADS;
  float* xc = xA;
  float* xn = xB;
  for (int k = 0; k < 3; ++k) {
    gather_step<<<gblocks, THREADS, 0, stream>>>(
        (const float4*)xc, (float4*)xn, (float4*)out,
        rowptr, csrc, cw, N, (k == 2) ? 0.25f : 1.0f);
    float* t = xc; xc = xn; xn = t;
  }
}